// FeatureFusion_3135326126765
// MI455X (gfx1250) — compile-verified
//
#include <hip/hip_runtime.h>
#include <math.h>

// ---------------------------------------------------------------------------
// FeatureFusion for MI455X (gfx1250, wave32).
// out[n,:] = sum_l wn[l,n] * p_l[n,:],  wn = softmax_n(p_l . W_l) cross-level
// normalized.  Biases cancel under softmax over n and are skipped (exact).
// Bandwidth-bound (~1.1 GB traffic @ 23.3 TB/s ~= 48us); logits pass uses
// V_WMMA_F32_16X16X4_F32 (fp32-exact) with B columns replicated.
// ---------------------------------------------------------------------------

typedef float v2f __attribute__((ext_vector_type(2)));
typedef float v4f __attribute__((ext_vector_type(4)));
typedef float v8f __attribute__((ext_vector_type(8)));

// Order-preserving float <-> uint mapping for atomicMax on floats.
__device__ __forceinline__ unsigned fmap(float f) {
    unsigned u = __float_as_uint(f);
    return (u & 0x80000000u) ? ~u : (u | 0x80000000u);
}
__device__ __forceinline__ float funmap(unsigned u) {
    unsigned v = (u & 0x80000000u) ? (u ^ 0x80000000u) : ~u;
    return __uint_as_float(v);
}

// ---------------------------------------------------------------------------
// K0: re-init reduction slots every launch (ws is only poisoned once).
// ---------------------------------------------------------------------------
__global__ void ff_init(unsigned* __restrict__ gmaxu, float* __restrict__ gsum) {
    int t = threadIdx.x;
    if (t < 5) {
        gmaxu[t] = 0x007FFFFFu;   // fmap(-inf)
        gsum[t]  = 0.0f;
    }
}

// ---------------------------------------------------------------------------
// K1: logits[l][n] = dot(p_l[n,:], W_l)  via V_WMMA_F32_16X16X4_F32.
// One wave = one 16-sample tile, accumulating 64 k-chunks of 4 per level.
// A (16x4 f32): lanes 0-15 -> {K0,K1}, lanes 16-31 -> {K2,K3} (ISA layout).
// B (4x16 f32): W chunk replicated across the 16 columns -> every C column
// holds the same logits; C vgpr g = row r0 + g + 8*(lane>=16).
// Also produces the global per-level max (softmax stability).
// ---------------------------------------------------------------------------
__global__ __launch_bounds__(256) void ff_logits(
    const float* __restrict__ p0, const float* __restrict__ p1,
    const float* __restrict__ p2, const float* __restrict__ p3,
    const float* __restrict__ p4,
    const float* __restrict__ w0, const float* __restrict__ w1,
    const float* __restrict__ w2, const float* __restrict__ w3,
    const float* __restrict__ w4,
    float* __restrict__ logits, unsigned* __restrict__ gmaxu,
    int N, int ntiles)
{
    __shared__ float ldsW[5 * 256];

    const float* P[5] = {p0, p1, p2, p3, p4};
    const float* W[5] = {w0, w1, w2, w3, w4};

    int t = threadIdx.x;
    // Stage the 5 weight vectors (5 KB) into LDS once per block.
#pragma unroll
    for (int l = 0; l < 5; ++l) ldsW[l * 256 + t] = W[l][t];
    __syncthreads();

    int lane = t & 31;
    int wave = t >> 5;
    int tile = blockIdx.x * 8 + wave;
    if (tile >= ntiles) return;            // wave-uniform: EXEC stays all-1s

    int r0  = tile * 16;
    int m   = lane & 15;                   // A-matrix row within tile
    int kh  = lane >> 4;                   // 0: K0/K1 half, 1: K2/K3 half
    int row = r0 + m; if (row >= N) row = N - 1;   // clamp (N%16==0 normally)
    size_t rowoff = (size_t)row * 256;

    v8f c[5] = {};                         // fp32 accumulators, one per level

#pragma unroll 2
    for (int k0 = 0; k0 < 256; k0 += 4) {
        int idx = k0 + kh * 2;
#pragma unroll
        for (int l = 0; l < 5; ++l) {
            v2f a = *(const v2f*)(P[l] + rowoff + idx);        // global_load_b64
            v2f b = *(const v2f*)(&ldsW[l * 256 + idx]);       // ds_load_b64
            // 8 args: (neg_a, A, neg_b, B, c_mod, C, reuse_a, reuse_b)
            c[l] = __builtin_amdgcn_wmma_f32_16x16x4_f32(
                false, a, false, b, (short)0, c[l], false, false);
        }
    }

#pragma unroll
    for (int l = 0; l < 5; ++l) {
        // Wave max for this level (columns are replicated -> per-lane C vgprs
        // already cover 8 rows; combine the two K-halves with one shuffle).
        float mx = c[l][0];
#pragma unroll
        for (int g = 1; g < 8; ++g) mx = fmaxf(mx, c[l][g]);
        mx = fmaxf(mx, __shfl_xor(mx, 16, 32));
        if (lane == 0) atomicMax(&gmaxu[l], fmap(mx));

        // Column 0 lanes (lane 0 and lane 16) own rows g and g+8.
        if ((lane & 15) == 0) {
#pragma unroll
            for (int g = 0; g < 8; ++g) {
                int rr = r0 + kh * 8 + g;
                if (rr < N) logits[(size_t)l * N + rr] = c[l][g];
            }
        }
    }
}

// ---------------------------------------------------------------------------
// K2: in-place exp(logit - max_l), wave-reduced atomicAdd into per-level sums.
// ---------------------------------------------------------------------------
__global__ __launch_bounds__(256) void ff_expsum(
    float* __restrict__ ev, const unsigned* __restrict__ gmaxu,
    float* __restrict__ gsum, int N)
{
    int n    = blockIdx.x * 256 + threadIdx.x;
    int lane = threadIdx.x & 31;
#pragma unroll
    for (int l = 0; l < 5; ++l) {
        float e = 0.0f;
        if (n < N) {
            float mx = funmap(gmaxu[l]);
            e = expf(ev[(size_t)l * N + n] - mx);
            ev[(size_t)l * N + n] = e;
        }
        float s = e;
#pragma unroll
        for (int off = 16; off > 0; off >>= 1) s += __shfl_xor(s, off, 32);
        if (lane == 0) atomicAdd(&gsum[l], s);
    }
}

// ---------------------------------------------------------------------------
// K3: one wave per sample; lane covers 8 of the 256 features.
// Streaming pass: non-temporal b128 loads (last use of p) + NT stores.
// ---------------------------------------------------------------------------
__global__ __launch_bounds__(256) void ff_combine(
    const float* __restrict__ p0, const float* __restrict__ p1,
    const float* __restrict__ p2, const float* __restrict__ p3,
    const float* __restrict__ p4,
    const float* __restrict__ ev, const float* __restrict__ gsum,
    float* __restrict__ out, int N)
{
    const float* P[5] = {p0, p1, p2, p3, p4};
    int lane = threadIdx.x & 31;
    int wave = threadIdx.x >> 5;
    int n = blockIdx.x * 8 + wave;
    if (n >= N) return;

    // Per-sample weights: w_l = e_l / sum_n e_l, then normalize across levels.
    float wl[5];
    float tot = 0.0f;
#pragma unroll
    for (int l = 0; l < 5; ++l) {
        float w = ev[(size_t)l * N + n] / gsum[l];
        wl[l] = w;
        tot += w;
    }
    float inv = 1.0f / tot;

    size_t base = (size_t)n * 256 + lane * 8;
    v4f acc0 = {};
    v4f acc1 = {};
#pragma unroll
    for (int l = 0; l < 5; ++l) {
        float w = wl[l] * inv;
        v4f a = __builtin_nontemporal_load((const v4f*)(P[l] + base));
        v4f b = __builtin_nontemporal_load((const v4f*)(P[l] + base + 4));
        acc0 += w * a;
        acc1 += w * b;
    }
    __builtin_nontemporal_store(acc0, (v4f*)(out + base));
    __builtin_nontemporal_store(acc1, (v4f*)(out + base + 4));
}

// ---------------------------------------------------------------------------
// Host launcher. Inputs: p3..p7, then W3,b3,W4,b4,...  (biases are unused:
// softmax over the batch axis is shift-invariant, so they cancel exactly).
// ws layout: [5*N floats: logits -> exp values][5 u32 max][5 f32 sum]
// ---------------------------------------------------------------------------
extern "C" void kernel_launch(void* const* d_in, const int* in_sizes, int n_in,
                              void* d_out, int out_size, void* d_ws, size_t ws_size,
                              hipStream_t stream)
{
    const float* p[5];
    const float* W[5];
    for (int i = 0; i < 5; ++i) p[i] = (const float*)d_in[i];
    for (int i = 0; i < 5; ++i) W[i] = (const float*)d_in[5 + 2 * i];

    const int D = 256;
    const int N = in_sizes[0] / D;

    float*    ws     = (float*)d_ws;
    float*    logits = ws;                                    // 5*N floats
    unsigned* gmaxu  = (unsigned*)(ws + (size_t)5 * N);       // 5 u32
    float*    gsum   = (float*)(gmaxu + 5);                   // 5 f32

    ff_init<<<1, 32, 0, stream>>>(gmaxu, gsum);

    int ntiles  = (N + 15) / 16;
    int blocks1 = (ntiles + 7) / 8;
    ff_logits<<<blocks1, 256, 0, stream>>>(
        p[0], p[1], p[2], p[3], p[4],
        W[0], W[1], W[2], W[3], W[4],
        logits, gmaxu, N, ntiles);

    int blocks2 = (N + 255) / 256;
    ff_expsum<<<blocks2, 256, 0, stream>>>(logits, gmaxu, gsum, N);

    int blocks3 = (N + 7) / 8;
    ff_combine<<<blocks3, 256, 0, stream>>>(
        p[0], p[1], p[2], p[3], p[4],
        logits, gsum, (float*)d_out, N);
}